// FEAf_lowpass_32057635897476
// MI455X (gfx1250) — compile-verified
//
#include <hip/hip_runtime.h>
#include <math.h>

typedef __attribute__((ext_vector_type(2))) float v2f;
typedef __attribute__((ext_vector_type(8))) float v8f;

#define T_LEN 2048
#define OMEGA 0.0030679615757712823f /* 2*pi/2048 */

__device__ __forceinline__ v8f wmma4(v2f a, v2f b, v8f c) {
  // V_WMMA_F32_16X16X4_F32: D(16x16 f32) = A(16x4) * B(4x16) + C
  return __builtin_amdgcn_wmma_f32_16x16x4_f32(false, a, false, b, (short)0, c,
                                               false, false);
}

// ---------------------------------------------------------------------------
// Kernel 1: forward DFT (16 lowest rfft modes) for q,k,v.
// One wave per (tensor, b, c, h-tile of 16). M rows = h (contiguous dim!),
// K = t (full 2048 reduction, 512 WMMA steps x2 for cos/sin accumulators).
// Twiddles generated by complex-rotation recurrence, reseeded every 64 steps.
// Output: Fre/Fim [tensor][b][h][c][m]  (m = 0..15)
// ---------------------------------------------------------------------------
__global__ __launch_bounds__(32) void k1_fwd(const float* __restrict__ q,
                                             const float* __restrict__ k,
                                             const float* __restrict__ v,
                                             float* __restrict__ Fre,
                                             float* __restrict__ Fim) {
  const int bid  = blockIdx.x;          // ((tsel*4 + b)*64 + c)*4 + ht
  const int ht   = bid & 3;
  const int c    = (bid >> 2) & 63;
  const int b    = (bid >> 8) & 3;
  const int tsel = bid >> 10;
  const float* __restrict__ src = (tsel == 0) ? q : (tsel == 1) ? k : v;

  const int lane = threadIdx.x;
  const int hi   = lane >> 4;   // half-wave select
  const int lo   = lane & 15;

  // A element: x[b, t, c, ht*16+lo]  (h is the fastest-varying dim -> 64B segs)
  const float* sb = src + (size_t)b * T_LEN * 4096 + (size_t)c * 64 + ht * 16 + lo;

  // B fragment: lane provides B[koff, n=lo] = trig(OMEGA * lo * (t0+koff)),
  // koff = 2*hi + {0,1}. Per-step rotation by OMEGA*lo*4.
  const float n   = (float)lo;
  const float dth = OMEGA * n * 4.0f;
  const float cd  = cosf(dth), sd = sinf(dth);

  v8f accC = {}; v8f accS = {};
  for (int tb = 0; tb < T_LEN; tb += 256) {   // reseed every 64 K-steps
    const float th0 = OMEGA * n * (float)(tb + 2 * hi);
    const float th1 = OMEGA * n * (float)(tb + 2 * hi + 1);
    float c0 = cosf(th0), s0 = sinf(th0);
    float c1 = cosf(th1), s1 = sinf(th1);
#pragma unroll 4
    for (int tt = 0; tt < 256; tt += 4) {
      const int t = tb + tt;
      v2f a;
      a.x = sb[(size_t)(t + 2 * hi) * 4096];
      a.y = sb[(size_t)(t + 2 * hi + 1) * 4096];
      v2f bc; bc.x = c0; bc.y = c1;
      v2f bs; bs.x = s0; bs.y = s1;
      accC = wmma4(a, bc, accC);
      accS = wmma4(a, bs, accS);
      const float n0 = c0 * cd - s0 * sd; s0 = s0 * cd + c0 * sd; c0 = n0;
      const float n1 = c1 * cd - s1 * sd; s1 = s1 * cd + c1 * sd; c1 = n1;
    }
  }
  // D layout: lane holds (M = hi*8 + r, N = lo). rfft: X = accC - i*accS.
  const size_t fb =
      ((((size_t)tsel * 4 + b) * 64 + (ht * 16 + hi * 8)) * 64 + c) * 16 + lo;
#pragma unroll
  for (int r = 0; r < 8; ++r) {
    Fre[fb + (size_t)r * 1024] = accC[r];   // h stride = 64*16
    Fim[fb + (size_t)r * 1024] = -accS[r];
  }
}

// robust complex tanh matching jnp.tanh on complex64
__device__ __forceinline__ void ctanhf2(float x, float y, float& tr, float& ti) {
  if (fabsf(x) > 20.0f) { tr = copysignf(1.0f, x); ti = 0.0f; return; }
  const float s2x = sinhf(2.0f * x);
  const float c2x = coshf(2.0f * x);
  const float s2y = sinf(2.0f * y);
  const float c2y = cosf(2.0f * y);
  const float d = c2x + c2y;
  tr = s2x / d;
  ti = s2y / d;
}

// ---------------------------------------------------------------------------
// Kernel 2: per (b,h) complex attention over the 16 modes.
// qk[x,y] = sum_d qf[d,x]*kf[d,y]  (complex, K=64) ; tanh ; qkv = vf * qk^T.
// One wave per (b,h); tanh(qk) re-fragmented through 2KB LDS.
// ---------------------------------------------------------------------------
__global__ __launch_bounds__(32) void k2_attn(const float* __restrict__ Fre,
                                              const float* __restrict__ Fim,
                                              float* __restrict__ Yre,
                                              float* __restrict__ Yim) {
  __shared__ float qkre[256], qkim[256];
  const int bh = blockIdx.x;                 // b*64 + h
  const int lane = threadIdx.x, hi = lane >> 4, lo = lane & 15;
  const size_t PL = (size_t)4 * 64 * 64 * 16;     // per-tensor plane
  const size_t base = (size_t)bh * 64 * 16;       // [b][h][d][m]
  const float* Qre = Fre + base;            const float* Qim = Fim + base;
  const float* Kre = Fre + PL + base;       const float* Kim = Fim + PL + base;
  const float* Vre = Fre + 2 * PL + base;   const float* Vim = Fim + 2 * PL + base;

  v8f ar = {}, ai = {};   // qk accumulators (re, im); M=x, N=y, K=d
#pragma unroll 4
  for (int d0 = 0; d0 < 64; d0 += 4) {
    const int dk = d0 + 2 * hi;
    v2f aqr  = { Qre[dk * 16 + lo], Qre[(dk + 1) * 16 + lo] };
    v2f aqi  = { Qim[dk * 16 + lo], Qim[(dk + 1) * 16 + lo] };
    v2f aqiN = { -aqi.x, -aqi.y };
    v2f bkr  = { Kre[dk * 16 + lo], Kre[(dk + 1) * 16 + lo] };
    v2f bki  = { Kim[dk * 16 + lo], Kim[(dk + 1) * 16 + lo] };
    ar = wmma4(aqr, bkr, ar);
    ar = wmma4(aqiN, bki, ar);   // Re = ReRe - ImIm
    ai = wmma4(aqr, bki, ai);
    ai = wmma4(aqi, bkr, ai);    // Im = ReIm + ImRe
  }
#pragma unroll
  for (int r = 0; r < 8; ++r) {
    float tr, ti;
    ctanhf2(ar[r], ai[r], tr, ti);
    const int x = hi * 8 + r;
    qkre[x * 16 + lo] = tr;
    qkim[x * 16 + lo] = ti;
  }
  __syncthreads();

  // qkv[d,x] = sum_y vf[d,y]*qk[x,y] : A = vf (M=d), B[k=y, n=x] = qk[x,y]
  for (int d0 = 0; d0 < 64; d0 += 16) {
    v8f cr = {}, ci = {};
#pragma unroll
    for (int s = 0; s < 4; ++s) {
      const int y0 = s * 4 + 2 * hi;
      const int vrow = (d0 + lo) * 16;
      v2f avr  = { Vre[vrow + y0], Vre[vrow + y0 + 1] };
      v2f avi  = { Vim[vrow + y0], Vim[vrow + y0 + 1] };
      v2f aviN = { -avi.x, -avi.y };
      v2f bqr  = { qkre[lo * 16 + y0], qkre[lo * 16 + y0 + 1] };
      v2f bqi  = { qkim[lo * 16 + y0], qkim[lo * 16 + y0 + 1] };
      cr = wmma4(avr, bqr, cr);
      cr = wmma4(aviN, bqi, cr);
      ci = wmma4(avr, bqi, ci);
      ci = wmma4(avi, bqr, ci);
    }
#pragma unroll
    for (int r = 0; r < 8; ++r) {
      const int d = d0 + hi * 8 + r;
      Yre[base + d * 16 + lo] = cr[r];
      Yim[base + d * 16 + lo] = ci[r];
    }
  }
}

// ---------------------------------------------------------------------------
// Kernel 3: inverse rfft (16 bins) as one K=32 matmul:
// out[h,t] = sum_{k<16} a_re[h,k]*cos(w_k t) + sum_{k<16} (-a_im[h,k])*sin(w_k t)
// One wave per (b, c, h-tile). A' constant across t; N = t tiles of 16 with
// rotating cos/sin B-fragments, reseeded every 512 t.
// ---------------------------------------------------------------------------
__global__ __launch_bounds__(32) void k3_inv(const float* __restrict__ Yre,
                                             const float* __restrict__ Yim,
                                             float* __restrict__ out) {
  const int bid = blockIdx.x;               // (b*64 + c)*4 + ht
  const int ht  = bid & 3;
  const int c   = (bid >> 2) & 63;
  const int b   = bid >> 8;
  const int lane = threadIdx.x, hi = lane >> 4, lo = lane & 15;

  // constant A' fragments: rows = h, K = [alpha*ReY (0..15) | -alpha*ImY (16..31)]
  const int h = ht * 16 + lo;
  const size_t yb = (((size_t)b * 64 + h) * 64 + c) * 16;
  v2f afr[8];
#pragma unroll
  for (int s = 0; s < 8; ++s) {
#pragma unroll
    for (int vv = 0; vv < 2; ++vv) {
      const int kk = s * 4 + 2 * hi + vv;
      const int m = kk & 15;
      const float alpha =
          ((m == 0) ? 1.0f : 2.0f) * (1.0f / (2048.0f * 4096.0f));
      const float val = (kk < 16) ? (Yre[yb + m] * alpha) : (-Yim[yb + m] * alpha);
      if (vv == 0) afr[s].x = val; else afr[s].y = val;
    }
  }

  // per-lane angle states for its 8 modes: m_j = 4*(j>>1) + 2*hi + (j&1)
  float stc[8], sts[8], rc[8], rs[8];
#pragma unroll
  for (int j = 0; j < 8; ++j) {
    const int m = (j >> 1) * 4 + 2 * hi + (j & 1);
    const float dA = OMEGA * (float)m * 16.0f;   // t advances 16 per tile
    rc[j] = cosf(dA); rs[j] = sinf(dA);
  }

  float* ob = out + (size_t)b * T_LEN * 4096 + (size_t)c * 64;
  for (int tb = 0; tb < T_LEN; tb += 512) {      // reseed every 32 tiles
#pragma unroll
    for (int j = 0; j < 8; ++j) {
      const int m = (j >> 1) * 4 + 2 * hi + (j & 1);
      const float th = OMEGA * (float)m * (float)(tb + lo);
      stc[j] = cosf(th); sts[j] = sinf(th);
    }
    for (int t0 = tb; t0 < tb + 512; t0 += 16) {
      v8f acc = {};
#pragma unroll
      for (int s = 0; s < 4; ++s) {              // cos half (k = 0..15)
        v2f bc = { stc[2 * s], stc[2 * s + 1] };
        acc = wmma4(afr[s], bc, acc);
      }
#pragma unroll
      for (int s = 0; s < 4; ++s) {              // sin half (k = 16..31)
        v2f bs = { sts[2 * s], sts[2 * s + 1] };
        acc = wmma4(afr[s + 4], bs, acc);
      }
#pragma unroll
      for (int r = 0; r < 8; ++r) {              // D: M = h row, N = t col
        const int hh = ht * 16 + hi * 8 + r;
        ob[(size_t)(t0 + lo) * 4096 + hh] = acc[r];
      }
#pragma unroll
      for (int j = 0; j < 8; ++j) {
        const float nc = stc[j] * rc[j] - sts[j] * rs[j];
        sts[j] = sts[j] * rc[j] + stc[j] * rs[j];
        stc[j] = nc;
      }
    }
  }
}

extern "C" void kernel_launch(void* const* d_in, const int* in_sizes, int n_in,
                              void* d_out, int out_size, void* d_ws,
                              size_t ws_size, hipStream_t stream) {
  (void)in_sizes; (void)n_in; (void)out_size; (void)ws_size;
  const float* q = (const float*)d_in[0];
  const float* k = (const float*)d_in[1];
  const float* v = (const float*)d_in[2];
  float* out = (float*)d_out;

  // workspace layout (floats): Fre[3][4][64][64][16], Fim same, Yre/Yim[4][64][64][16]
  float* w   = (float*)d_ws;
  float* Fre = w;                         // 786432 floats
  float* Fim = Fre + 786432;              // 786432 floats
  float* Yre = Fim + 786432;              // 262144 floats
  float* Yim = Yre + 262144;              // 262144 floats  (total 8 MB)

  dim3 blk(32);
  k1_fwd<<<3072, blk, 0, stream>>>(q, k, v, Fre, Fim);
  k2_attn<<<256, blk, 0, stream>>>(Fre, Fim, Yre, Yim);
  k3_inv<<<1024, blk, 0, stream>>>(Yre, Yim, out);
}